// _PointNet2TfXModel_6227702580012
// MI455X (gfx1250) — compile-verified
//
#include <hip/hip_runtime.h>

// PointNet++ (4×SA + 4×FP + conv) forward for MI455X / gfx1250.
// Dense einsums on V_WMMA_F32_16X16X4_F32 with padded/packed operands so the
// GEMM inner loop is branch-free: 3 b64 loads + 2 WMMAs per k-step.

#define BATCH   4
#define NPTS    8192
#define NSAMPLE 32
#define NCLASS  13

typedef __attribute__((ext_vector_type(2))) float v2f;
typedef __attribute__((ext_vector_type(8))) float v8f;

// ---------------------------------------------------------------------------
// Weight packing: Wp[(k>>1)*Np + n] = { W[k][n], W[k+1][n] }, zero padded to
// Kp (mult of 4) x Np (mult of 32).  This is exactly the per-lane B fragment
// of V_WMMA_F32_16X16X4_F32, so the GEMM loads it with one global_load_b64.
// ---------------------------------------------------------------------------
__global__ void pack_w_kernel(const float* __restrict__ W, v2f* __restrict__ Wp,
                              int K, int N, int Kp, int Np)
{
    size_t total  = (size_t)(Kp >> 1) * Np;
    size_t stride = (size_t)gridDim.x * blockDim.x;
    for (size_t e = (size_t)blockIdx.x * blockDim.x + threadIdx.x; e < total; e += stride) {
        int kk = (int)(e / Np);
        int n  = (int)(e - (size_t)kk * Np);
        v2f v;
        v.x = (2 * kk     < K && n < N) ? W[(size_t)(2 * kk) * N + n]     : 0.f;
        v.y = (2 * kk + 1 < K && n < N) ? W[(size_t)(2 * kk + 1) * N + n] : 0.f;
        Wp[e] = v;
    }
}

// ---------------------------------------------------------------------------
// Branch-free WMMA f32 GEMM: Out[M,N] = act( A[M,Kp(pad0)] * W[Kp,N] + bias )
// Preconditions: M % 16 == 0, Kp % 4 == 0 (A zero-padded to Kp, row stride Kp),
// Wp packed/padded to Kp x Np (Np % 32 == 0).  One wave -> 16x32 output tile,
// two accumulator chains sharing one A fragment, 1-deep load pipeline.
// f32 16x16x4 lane layout (ISA 7.12.2):
//   A: lane l holds A[m0+(l&15)][k + 2*(l>>4) + {0,1}]
//   B: lane l holds B[k + 2*(l>>4) + {0,1}][n0+(l&15)]   (== packed float2)
//   D: vgpr i, lane l -> D[m0 + i + 8*(l>>4)][n0 + (l&15)]
// ---------------------------------------------------------------------------
__global__ void gemm_wmma_f32_kernel(const float* __restrict__ A, int Kp,
                                     const v2f* __restrict__ Wp, int Np,
                                     float* __restrict__ Out,
                                     int M, int N,
                                     int relu, const float* __restrict__ bias)
{
    const int wavesPerBlock = blockDim.x >> 5;
    int wave    = blockIdx.x * wavesPerBlock + ((int)threadIdx.x >> 5);
    int lane    = threadIdx.x & 31;
    int tilesN2 = Np >> 5;                 // pairs of 16-col tiles
    int tilesM  = M >> 4;
    int total   = tilesM * tilesN2;
    int stride  = gridDim.x * wavesPerBlock;
    int r  = lane & 15;
    int hi = lane >> 4;

    for (int t = wave; t < total; t += stride) {
        int tm = t / tilesN2, tn = t - tm * tilesN2;
        int m0 = tm << 4, n0 = tn << 5;

        const float* Arow = A  + (size_t)(m0 + r) * Kp + (hi << 1);
        const v2f*   W0   = Wp + (size_t)hi * Np + (n0 + r);
        const v2f*   W1   = W0 + 16;

        v8f acc0 = {0.f,0.f,0.f,0.f,0.f,0.f,0.f,0.f};
        v8f acc1 = {0.f,0.f,0.f,0.f,0.f,0.f,0.f,0.f};

        // 1-deep software pipeline: loads for step k+4 issue before WMMAs of k.
        v2f a  = *(const v2f*)Arow;
        v2f b0 = W0[0];
        v2f b1 = W1[0];
        for (int k = 4; k < Kp; k += 4) {
            v2f an  = *(const v2f*)(Arow + k);
            v2f bn0 = W0[(size_t)(k >> 1) * Np];
            v2f bn1 = W1[(size_t)(k >> 1) * Np];
            acc0 = __builtin_amdgcn_wmma_f32_16x16x4_f32(false, a, false, b0,
                                                         (short)0, acc0, false, false);
            acc1 = __builtin_amdgcn_wmma_f32_16x16x4_f32(false, a, false, b1,
                                                         (short)0, acc1, false, false);
            a = an; b0 = bn0; b1 = bn1;
        }
        acc0 = __builtin_amdgcn_wmma_f32_16x16x4_f32(false, a, false, b0,
                                                     (short)0, acc0, false, false);
        acc1 = __builtin_amdgcn_wmma_f32_16x16x4_f32(false, a, false, b1,
                                                     (short)0, acc1, false, false);

        int nn0 = n0 + r;
        int nn1 = n0 + 16 + r;
#pragma unroll
        for (int i = 0; i < 8; ++i) {
            int mm = m0 + i + (hi << 3);          // always < M (M % 16 == 0)
            if (nn0 < N) {
                float v = acc0[i];
                if (bias) v += bias[nn0];
                if (relu) v = fmaxf(v, 0.f);
                Out[(size_t)mm * N + nn0] = v;
            }
            if (nn1 < N) {
                float v = acc1[i];
                if (bias) v += bias[nn1];
                if (relu) v = fmaxf(v, 0.f);
                Out[(size_t)mm * N + nn1] = v;
            }
        }
    }
}

// ---------------------------------------------------------------------------
// SA neighbor selection. One 256-thread workgroup per (b, center).
// Replicates: argsort by distance (stable, tie -> lower index), take 32,
// valid = d < radius, valid indices sorted ascending, invalid filled with
// smallest valid index. FPS permutation approximated by coprime stride.
// ---------------------------------------------------------------------------
__global__ void sa_neighbors_kernel(const float* __restrict__ xyz,   // [B,n,3]
                                    float* __restrict__ new_xyz,     // [B,np,3]
                                    int*   __restrict__ idx_out,     // [B,np,NSAMPLE]
                                    int n, int npoint, float radius)
{
    __shared__ float sdist[8192];
    __shared__ float rmin[256];
    __shared__ int   ridx[256];
    __shared__ int   ssel[NSAMPLE];
    __shared__ float sseld[NSAMPLE];

    int b   = blockIdx.x / npoint;
    int j   = blockIdx.x % npoint;
    int tid = threadIdx.x;

    int fpsj = (int)(((long long)j * 2531LL) % n);   // deterministic pseudo-FPS
    float cx = xyz[((size_t)b * n + fpsj) * 3 + 0];
    float cy = xyz[((size_t)b * n + fpsj) * 3 + 1];
    float cz = xyz[((size_t)b * n + fpsj) * 3 + 2];
    if (tid == 0) {
        new_xyz[((size_t)b * npoint + j) * 3 + 0] = cx;
        new_xyz[((size_t)b * npoint + j) * 3 + 1] = cy;
        new_xyz[((size_t)b * npoint + j) * 3 + 2] = cz;
    }

    for (int i = tid; i < n; i += 256) {
        float dx = xyz[((size_t)b * n + i) * 3 + 0] - cx;
        float dy = xyz[((size_t)b * n + i) * 3 + 1] - cy;
        float dz = xyz[((size_t)b * n + i) * 3 + 2] - cz;
        sdist[i] = sqrtf(dx * dx + dy * dy + dz * dz + 1e-12f);
    }
    __syncthreads();

    for (int s = 0; s < NSAMPLE; ++s) {
        float best = __builtin_inff();
        int   bi   = 0x7fffffff;
        for (int i = tid; i < n; i += 256) {
            float d = sdist[i];
            if (d < best || (d == best && i < bi)) { best = d; bi = i; }
        }
        rmin[tid] = best; ridx[tid] = bi;
        __syncthreads();
        for (int off = 128; off > 0; off >>= 1) {
            if (tid < off) {
                float dv = rmin[tid + off]; int iv = ridx[tid + off];
                if (dv < rmin[tid] || (dv == rmin[tid] && iv < ridx[tid])) {
                    rmin[tid] = dv; ridx[tid] = iv;
                }
            }
            __syncthreads();
        }
        if (tid == 0) {
            ssel[s]  = ridx[0];
            sseld[s] = rmin[0];
            sdist[ridx[0]] = __builtin_inff();
        }
        __syncthreads();
    }

    if (tid == 0) {
        int tmp[NSAMPLE];
        int cnt = 0;
        for (int s = 0; s < NSAMPLE; ++s)
            if (sseld[s] < radius) tmp[cnt++] = ssel[s];
        for (int a = 1; a < cnt; ++a) {          // ascending insertion sort
            int v = tmp[a], p = a - 1;
            while (p >= 0 && tmp[p] > v) { tmp[p + 1] = tmp[p]; --p; }
            tmp[p + 1] = v;
        }
        int fill = (cnt > 0) ? tmp[0] : fpsj;
        int* o = idx_out + ((size_t)b * npoint + j) * NSAMPLE;
        for (int s = 0; s < NSAMPLE; ++s) o[s] = (s < cnt) ? tmp[s] : fill;
    }
}

// Gather + relative-coordinate concat into zero-padded rows of width cinp.
__global__ void sa_gather_kernel(const float* __restrict__ xyz,
                                 const float* __restrict__ pts,
                                 const float* __restrict__ new_xyz,
                                 const int*   __restrict__ idx,
                                 float* __restrict__ A,
                                 int n, int c, int npoint, int cinp)
{
    int cin = 3 + c;
    size_t total  = (size_t)BATCH * npoint * NSAMPLE * cinp;
    size_t stride = (size_t)gridDim.x * blockDim.x;
    for (size_t e = (size_t)blockIdx.x * blockDim.x + threadIdx.x; e < total; e += stride) {
        size_t row = e / cinp;
        int ch = (int)(e - row * cinp);
        float v = 0.f;
        if (ch < cin) {
            size_t bj = row / NSAMPLE;
            int jj = (int)(bj % npoint);
            int bb = (int)(bj / npoint);
            int src = idx[row];
            if (ch < 3)
                v = xyz[((size_t)bb * n + src) * 3 + ch]
                  - new_xyz[((size_t)bb * npoint + jj) * 3 + ch];
            else
                v = pts[((size_t)bb * n + src) * c + (ch - 3)];
        }
        A[e] = v;
    }
}

// Max over the NSAMPLE axis: out[g, f] = max_s T[(g*NSAMPLE+s), f]
__global__ void maxpool_kernel(const float* __restrict__ T,
                               float* __restrict__ out, int groups, int F)
{
    size_t total  = (size_t)groups * F;
    size_t stride = (size_t)gridDim.x * blockDim.x;
    for (size_t e = (size_t)blockIdx.x * blockDim.x + threadIdx.x; e < total; e += stride) {
        size_t g = e / F;
        int f = (int)(e - g * F);
        float m = -__builtin_inff();
        for (int s = 0; s < NSAMPLE; ++s)
            m = fmaxf(m, T[((size_t)g * NSAMPLE + s) * F + f]);
        out[e] = m;
    }
}

// 3-NN with inverse-distance weights (reference semantics, incl. 1/(1+1e-6))
__global__ void three_nn_kernel(const float* __restrict__ tx,   // [B,nt,3]
                                const float* __restrict__ sx,   // [B,ns,3]
                                int nt, int ns,
                                int* __restrict__ ni, float* __restrict__ nw)
{
    size_t total  = (size_t)BATCH * nt;
    size_t stride = (size_t)gridDim.x * blockDim.x;
    for (size_t e = (size_t)blockIdx.x * blockDim.x + threadIdx.x; e < total; e += stride) {
        int b = (int)(e / nt);
        float px = tx[e * 3 + 0], py = tx[e * 3 + 1], pz = tx[e * 3 + 2];
        float d0 = __builtin_inff(), d1 = d0, d2 = d0;
        int   i0 = 0, i1 = 0, i2 = 0;
        for (int jj = 0; jj < ns; ++jj) {
            float dx = sx[((size_t)b * ns + jj) * 3 + 0] - px;
            float dy = sx[((size_t)b * ns + jj) * 3 + 1] - py;
            float dz = sx[((size_t)b * ns + jj) * 3 + 2] - pz;
            float d = sqrtf(dx * dx + dy * dy + dz * dz + 1e-12f);
            if (d < d0)      { d2 = d1; i2 = i1; d1 = d0; i1 = i0; d0 = d; i0 = jj; }
            else if (d < d1) { d2 = d1; i2 = i1; d1 = d;  i1 = jj; }
            else if (d < d2) { d2 = d;  i2 = jj; }
        }
        float n0 = fmaxf(d0, 1e-10f), n1 = fmaxf(d1, 1e-10f), n2 = fmaxf(d2, 1e-10f);
        float v0 = 1.f / n0, v1 = 1.f / n1, v2 = 1.f / n2;
        float inv = 1.f / ((v0 + v1 + v2) * (1.f + 1e-6f));
        ni[e * 3 + 0] = i0; ni[e * 3 + 1] = i1; ni[e * 3 + 2] = i2;
        nw[e * 3 + 0] = v0 * inv; nw[e * 3 + 1] = v1 * inv; nw[e * 3 + 2] = v2 * inv;
    }
}

// A[B*nt, cinp(pad0)] = [ interp(feat_s) | feat_t | 0 pad ]
__global__ void interp_concat_kernel(const float* __restrict__ fs,  // [B,ns,cs]
                                     const float* __restrict__ ft,  // [B,nt,ct]
                                     const int*   __restrict__ ni,
                                     const float* __restrict__ nw,
                                     float* __restrict__ A,
                                     int nt, int ns, int cs, int ct, int cinp)
{
    int cin = cs + ct;
    size_t total  = (size_t)BATCH * nt * cinp;
    size_t stride = (size_t)gridDim.x * blockDim.x;
    for (size_t e = (size_t)blockIdx.x * blockDim.x + threadIdx.x; e < total; e += stride) {
        size_t row = e / cinp;
        int ch = (int)(e - row * cinp);
        int b  = (int)(row / nt);
        float v = 0.f;
        if (ch < cs) {
            float acc = 0.f;
            for (int k = 0; k < 3; ++k) {
                int s = ni[row * 3 + k];
                acc += nw[row * 3 + k] * fs[((size_t)b * ns + s) * cs + ch];
            }
            v = acc;
        } else if (ch < cin) {
            v = ft[row * ct + (ch - cs)];
        }
        A[e] = v;
    }
}

// ---------------------------------------------------------------------------
static void launch_gemm(const float* A, int Kp, const v2f* Wp, int Np,
                        float* Out, int M, int N, int relu, const float* bias,
                        hipStream_t s)
{
    int waves  = (M >> 4) * (Np >> 5);
    int blocks = (waves + 7) / 8;                 // 8 waves (256 threads)/block
    gemm_wmma_f32_kernel<<<blocks, 256, 0, s>>>(A, Kp, Wp, Np, Out, M, N, relu, bias);
}

extern "C" void kernel_launch(void* const* d_in, const int* in_sizes, int n_in,
                              void* d_out, int out_size, void* d_ws, size_t ws_size,
                              hipStream_t stream)
{
    (void)in_sizes; (void)n_in; (void)out_size; (void)ws_size;

    // setup_inputs() insertion order: xyz, points, sa0(3), sa1(3), sa2(3),
    // sa3(3), fp0(2), fp1(2), fp2(2), fp3(3), conv_w, conv_b  -> 25 inputs
    const float* xyz    = (const float*)d_in[0];
    const float* points = (const float*)d_in[1];
    const float* Wsa[12];
    for (int i = 0; i < 12; ++i) Wsa[i] = (const float*)d_in[2 + i];
    const float* Wfp[9];
    for (int i = 0; i < 9; ++i)  Wfp[i] = (const float*)d_in[14 + i];
    const float* conv_w = (const float*)d_in[23];
    const float* conv_b = (const float*)d_in[24];

    float* base = (float*)d_ws;
    size_t off = 0;
    auto alloc = [&](size_t nf) { float* p = base + off; off += nf; return p; };

    float* xs1 = alloc((size_t)BATCH * 512 * 3);
    float* xs2 = alloc((size_t)BATCH * 128 * 3);
    float* xs3 = alloc((size_t)BATCH * 32 * 3);
    float* xs4 = alloc((size_t)BATCH * 8 * 3);
    float* ps1 = alloc((size_t)BATCH * 512 * 256);
    float* ps2 = alloc((size_t)BATCH * 128 * 128);
    float* ps3 = alloc((size_t)BATCH * 32 * 256);
    float* ps4 = alloc((size_t)BATCH * 8 * 512);
    int*   idxbuf = (int*)alloc((size_t)BATCH * 512 * NSAMPLE);
    int*   nibuf  = (int*)alloc((size_t)BATCH * NPTS * 3);
    float* nwbuf  = alloc((size_t)BATCH * NPTS * 3);
    float* buf0 = alloc((size_t)65536 * 512);   // SA0 intermediate, 134 MB
    float* buf1 = alloc((size_t)65536 * 512);

    // Pack a weight into WMMA-B fragment order, zero-padded to Kp x Np.
    auto pack = [&](const float* W, int K, int N, int& KpO, int& NpO) -> const v2f* {
        int Kp = (K + 3) & ~3;
        int Np = (N + 31) & ~31;
        v2f* dst = (v2f*)alloc((size_t)Kp * Np);        // (Kp/2)*Np float2s
        size_t tot = (size_t)(Kp >> 1) * Np;
        pack_w_kernel<<<(int)((tot + 255) / 256), 256, 0, stream>>>(W, dst, K, N, Kp, Np);
        KpO = Kp; NpO = Np;
        return dst;
    };

    // ---------------- SA stack ----------------
    const int   sa_np[4]    = {512, 128, 32, 8};
    const float sa_r [4]    = {0.1f, 0.2f, 0.4f, 0.8f};
    const int   sa_f [4][3] = {{512,512,256},{256,256,128},{128,128,256},{256,256,512}};
    float* xs_out[4] = {xs1, xs2, xs3, xs4};
    float* ps_out[4] = {ps1, ps2, ps3, ps4};

    const float* cx = xyz;    int n_src = NPTS;
    const float* cp = points; int c_src = 6;
    for (int li = 0; li < 4; ++li) {
        int np = sa_np[li];
        sa_neighbors_kernel<<<BATCH * np, 256, 0, stream>>>(
            cx, xs_out[li], idxbuf, n_src, np, sa_r[li]);

        int cin  = 3 + c_src;
        int cinp = (cin + 3) & ~3;
        size_t M = (size_t)BATCH * np * NSAMPLE;    // multiple of 16
        {
            size_t tot = M * cinp;
            sa_gather_kernel<<<(int)((tot + 255) / 256), 256, 0, stream>>>(
                cx, cp, xs_out[li], idxbuf, buf0, n_src, c_src, np, cinp);
        }
        float* in = buf0; float* outb = buf1;
        int Kact = cin, Kp = cinp;
        for (int l = 0; l < 3; ++l) {
            int N = sa_f[li][l];
            int KpW, NpW;
            const v2f* Wp = pack(Wsa[li * 3 + l], Kact, N, KpW, NpW);
            launch_gemm(in, Kp, Wp, NpW, outb, (int)M, N, 1, nullptr, stream);
            float* t = in; in = outb; outb = t;
            Kact = N; Kp = N;                       // N is a multiple of 32
        }
        {
            size_t tot = (size_t)BATCH * np * Kact;
            maxpool_kernel<<<(int)((tot + 255) / 256), 256, 0, stream>>>(
                in, ps_out[li], BATCH * np, Kact);
        }
        cx = xs_out[li]; n_src = np;
        cp = ps_out[li]; c_src = Kact;
    }

    // ---------------- FP stack ----------------
    const float* src_x[4] = {xs4, xs3, xs2, xs1};
    const int    src_n[4] = {8, 32, 128, 512};
    const float* tgt_x[4] = {xs3, xs2, xs1, xyz};
    const int    tgt_n[4] = {32, 128, 512, NPTS};
    const float* tgt_f[4] = {ps3, ps2, ps1, points};
    const int    tgt_c[4] = {256, 128, 256, 6};
    const int    fp_nl[4] = {2, 2, 2, 3};
    const int    fp_f[4][3] = {{256,256,0},{256,256,0},{256,128,0},{128,128,128}};

    const float* f = ps4; int cs = 512;
    int wi = 0;
    for (int li = 0; li < 4; ++li) {
        int nt = tgt_n[li], ns = src_n[li], ct = tgt_c[li];
        {
            size_t tot = (size_t)BATCH * nt;
            three_nn_kernel<<<(int)((tot + 255) / 256), 256, 0, stream>>>(
                tgt_x[li], src_x[li], nt, ns, nibuf, nwbuf);
        }
        float* A  = (f == buf0) ? buf1 : buf0;
        float* Bb = (A == buf0) ? buf1 : buf0;
        int cin  = cs + ct;
        int cinp = (cin + 3) & ~3;
        {
            size_t tot = (size_t)BATCH * nt * cinp;
            interp_concat_kernel<<<(int)((tot + 255) / 256), 256, 0, stream>>>(
                f, tgt_f[li], nibuf, nwbuf, A, nt, ns, cs, ct, cinp);
        }
        float* in = A; float* outb = Bb;
        int Kact = cin, Kp = cinp;
        for (int l = 0; l < fp_nl[li]; ++l) {
            int N = fp_f[li][l];
            int KpW, NpW;
            const v2f* Wp = pack(Wfp[wi + l], Kact, N, KpW, NpW);
            launch_gemm(in, Kp, Wp, NpW, outb, BATCH * nt, N, 1, nullptr, stream);
            float* t = in; in = outb; outb = t;
            Kact = N; Kp = N;
        }
        wi += fp_nl[li];
        f = in; cs = Kact;
    }

    // ---------------- final 128 -> 13 conv (+bias, no relu) ----------------
    {
        int KpW, NpW;
        const v2f* Wp = pack(conv_w, 128, NCLASS, KpW, NpW);   // Np = 32
        launch_gemm(f, 128, Wp, NpW, (float*)d_out, BATCH * NPTS, NCLASS, 0, conv_b, stream);
    }
}